// MMConv_7026566496850
// MI455X (gfx1250) — compile-verified
//
#include <hip/hip_runtime.h>
#include <math.h>

typedef __attribute__((ext_vector_type(2))) float v2f;
typedef __attribute__((ext_vector_type(8))) float v8f;

#define DD 64
#define ALPHA 0.1f
#define BETA 0.1f
#define THETA 0.4054651081081644f   // log(LAMDA/L + 1) = log(1.5)
#define EPSM 1e-16f

// ---------------- Kernel A: zero the moment/agg accumulators ----------------
__global__ void zero_ws_kernel(float* ws, long n) {
    long i = (long)blockIdx.x * blockDim.x + threadIdx.x;
    long stride = (long)gridDim.x * blockDim.x;
    for (; i < n; i += stride) ws[i] = 0.0f;
}

// ---------------- Kernel B: edge scatter (4 SpMMs, one h0 gather) -----------
#define PREFETCH_EDGES 2048   // lookahead distance (in edges)

__global__ __launch_bounds__(256)
void edge_scatter_kernel(const int* __restrict__ rows, const int* __restrict__ cols,
                         const float* __restrict__ vals,
                         const float* __restrict__ x, const float* __restrict__ h0,
                         float* aggx, float* mu, float* m2, float* m3, int E) {
    long t = (long)blockIdx.x * blockDim.x + threadIdx.x;
    int e = (int)(t >> 6);     // 64 lanes per edge (one per feature)
    int d = (int)(t & 63);
    if (e >= E) return;

    // latency-hiding prefetch of a future edge's source rows into cache
    int ep = e + PREFETCH_EDGES;
    if (ep < E) {
        int cp = cols[ep];
        __builtin_prefetch(&x[(size_t)cp * DD + d], 0, 1);
        __builtin_prefetch(&h0[(size_t)cp * DD + d], 0, 1);
    }

    int r = rows[e];
    int c = cols[e];
    float v = vals[e];
    float xv = x[c * DD + d];
    float hv = h0[c * DD + d];
    int o = r * DD + d;
    float hv2 = hv * hv;
    atomicAdd(&aggx[o], v * xv);
    atomicAdd(&mu[o],   v * hv);
    atomicAdd(&m2[o],   v * hv2);
    atomicAdd(&m3[o],   v * hv2 * hv);
}

// ------ Kernel B2: in-place moment transforms (sigma = sqrt, gamma = cbrt) --
__global__ void transform_moments_kernel(float* m2, float* m3, long n) {
    long i = (long)blockIdx.x * blockDim.x + threadIdx.x;
    long stride = (long)gridDim.x * blockDim.x;
    for (; i < n; i += stride) {
        float s = m2[i];
        s = (s == 0.0f) ? EPSM : s;
        m2[i] = sqrtf(s);
        float g = m3[i];
        g = (g == 0.0f) ? EPSM : g;
        m3[i] = cbrtf(g);          // signed cube root, matches reference
    }
}

// ---------------- Kernel C: fused node pass with f32 WMMA -------------------
__device__ __forceinline__ float eluf(float v) {
    return v > 0.0f ? v : expm1f(v);
}

#define WAVES_PER_BLOCK 2

__global__ __launch_bounds__(64)
void fused_node_kernel(const float* __restrict__ h0,
                       const float* __restrict__ weight,   // [64,64] row-major
                       const float* __restrict__ w_att,    // [128,64] row-major
                       const float* __restrict__ aggx,
                       const float* __restrict__ mu,       // k0 (transformed planes)
                       const float* __restrict__ m2,       // k1 = sigma
                       const float* __restrict__ m3,       // k2 = gamma
                       float* __restrict__ out) {
    __shared__ float sW[DD * DD];          // 16 KB
    __shared__ float sWatt[2 * DD * DD];   // 32 KB
    __shared__ float sT[WAVES_PER_BLOCK][16 * DD];  // per-wave 16x64 tile, 8 KB

    const int tid = threadIdx.x;
    for (int i = tid; i < DD * DD; i += 64)      sW[i]    = weight[i];
    for (int i = tid; i < 2 * DD * DD; i += 64)  sWatt[i] = w_att[i];
    __syncthreads();

    const int wave = tid >> 5;
    const int l    = tid & 31;
    const int half = l >> 4;       // 0: lanes 0-15, 1: lanes 16-31
    const int mrow = l & 15;       // A-matrix row for this lane
    const int ncol = l & 15;       // B/C/D column-in-tile for this lane
    const int r0   = (blockIdx.x * WAVES_PER_BLOCK + wave) * 16;
    float* T = sT[wave];

    // ---- stage h_agg = (1-alpha)*A·x + alpha*h0 into the wave tile ----
    for (int i = l; i < 16 * DD; i += 32) {
        int row = i >> 6, col = i & 63;
        int g = (r0 + row) * DD + col;
        T[i] = (1.0f - ALPHA) * aggx[g] + ALPHA * h0[g];
    }
    __syncthreads();

    // ---- GEMM1: hi = theta*(agg @ W) + (1-theta)*agg  (16x64 @ 64x64) ----
    v8f hi[4];
#pragma unroll
    for (int nt = 0; nt < 4; ++nt) {
        v8f acc = {};
        const int col = nt * 16 + ncol;
#pragma unroll
        for (int kk = 0; kk < 16; ++kk) {
            const int kb = kk * 4 + 2 * half;
            v2f a; a.x = T[mrow * DD + kb];       a.y = T[mrow * DD + kb + 1];
            v2f b; b.x = sW[kb * DD + col];       b.y = sW[(kb + 1) * DD + col];
            acc = __builtin_amdgcn_wmma_f32_16x16x4_f32(
                false, a, false, b, (short)0, acc, false, false);
        }
        hi[nt] = acc;
    }
#pragma unroll
    for (int nt = 0; nt < 4; ++nt) {
#pragma unroll
        for (int v = 0; v < 8; ++v) {
            int row = v + 8 * half;
            int col = nt * 16 + ncol;
            float agg = T[row * DD + col];
            hi[nt][v] = THETA * hi[nt][v] + (1.0f - THETA) * agg;
        }
    }
    __syncthreads();
    // overwrite tile with hi (full tile needed as A-matrix for the q-GEMM)
#pragma unroll
    for (int nt = 0; nt < 4; ++nt)
#pragma unroll
        for (int v = 0; v < 8; ++v)
            T[(v + 8 * half) * DD + nt * 16 + ncol] = hi[nt][v];
    __syncthreads();

    const float* __restrict__ kplane[3] = { mu, m2, m3 };

    // ---- attention:  e_m = elu(k_m @ Watt_top + hi @ Watt_bot) ----
#pragma unroll 1
    for (int nt = 0; nt < 4; ++nt) {
        const int col = nt * 16 + ncol;

        // q = hi @ Watt[64:128]  (shared across the 3 moments)
        v8f q = {};
#pragma unroll
        for (int kk = 0; kk < 16; ++kk) {
            const int kb = kk * 4 + 2 * half;
            v2f a; a.x = T[mrow * DD + kb];               a.y = T[mrow * DD + kb + 1];
            v2f b; b.x = sWatt[(DD + kb) * DD + col];     b.y = sWatt[(DD + kb + 1) * DD + col];
            q = __builtin_amdgcn_wmma_f32_16x16x4_f32(
                false, a, false, b, (short)0, q, false, false);
        }

        v8f e[3];
#pragma unroll 1
        for (int m = 0; m < 3; ++m) {
            const float* __restrict__ kp = kplane[m];
            v8f acc = q;   // start from the shared hi-half
#pragma unroll
            for (int kk = 0; kk < 16; ++kk) {
                const int kb = kk * 4 + 2 * half;                 // always even
                const int gi = (r0 + mrow) * DD + kb;             // 8B-aligned
                v2f a = *(const v2f*)(kp + gi);                   // global_load_b64
                v2f b; b.x = sWatt[kb * DD + col];
                       b.y = sWatt[(kb + 1) * DD + col];
                acc = __builtin_amdgcn_wmma_f32_16x16x4_f32(
                    false, a, false, b, (short)0, acc, false, false);
            }
            e[m] = acc;
        }

        // elementwise: elu -> softmax over moments -> blend, write out
#pragma unroll
        for (int v = 0; v < 8; ++v) {
            int row = v + 8 * half;
            int gi = (r0 + row) * DD + col;
            float e0 = eluf(e[0][v]);
            float e1 = eluf(e[1][v]);
            float e2 = eluf(e[2][v]);
            float mx = fmaxf(e0, fmaxf(e1, e2));
            float a0 = __expf(e0 - mx);
            float a1 = __expf(e1 - mx);
            float a2 = __expf(e2 - mx);
            float inv = 1.0f / (a0 + a1 + a2);
            float hm = (mu[gi] * a0 + m2[gi] * a1 + m3[gi] * a2) * inv;
            float hiv = T[row * DD + col];
            out[gi] = (1.0f - BETA) * hiv + BETA * hm;
        }
    }
}

// ---------------------------------------------------------------------------
extern "C" void kernel_launch(void* const* d_in, const int* in_sizes, int n_in,
                              void* d_out, int out_size, void* d_ws, size_t ws_size,
                              hipStream_t stream) {
    const float* x      = (const float*)d_in[0];
    const float* h0     = (const float*)d_in[1];
    const int*   rows   = (const int*)d_in[2];
    const int*   cols   = (const int*)d_in[3];
    const float* vals   = (const float*)d_in[4];
    const float* weight = (const float*)d_in[5];
    const float* w_att  = (const float*)d_in[6];
    float* out = (float*)d_out;

    const int N = in_sizes[0] / DD;
    const int E = in_sizes[2];

    float* ws   = (float*)d_ws;
    size_t plane = (size_t)N * DD;
    float* aggx = ws;
    float* mu   = ws + plane;
    float* m2v  = ws + 2 * plane;
    float* m3v  = ws + 3 * plane;

    // A: zero accumulators (4 planes of [N,64])
    zero_ws_kernel<<<2048, 256, 0, stream>>>(ws, (long)(4 * plane));

    // B: edge scatter, 64 lanes per edge
    long ethreads = (long)E * 64;
    int eblocks = (int)((ethreads + 255) / 256);
    edge_scatter_kernel<<<eblocks, 256, 0, stream>>>(rows, cols, vals, x, h0,
                                                     aggx, mu, m2v, m3v, E);

    // B2: in-place sigma = sqrt(m2), gamma = cbrt(m3)
    transform_moments_kernel<<<2048, 256, 0, stream>>>(m2v, m3v, (long)plane);

    // C: fused node pass, 32 rows per 64-thread block (N=65536 -> 2048 blocks)
    int nblocks = (N + 16 * WAVES_PER_BLOCK - 1) / (16 * WAVES_PER_BLOCK);
    fused_node_kernel<<<nblocks, 64, 0, stream>>>(h0, weight, w_att,
                                                  aggx, mu, m2v, m3v, out);
}